// MultiConv_21474836480481
// MI455X (gfx1250) — compile-verified
//
#include <hip/hip_runtime.h>
#include <hip/hip_bf16.h>
#include <stddef.h>

// ---------------------------------------------------------------------------
// MultiConv: 4-branch { Conv1x1(WMMA fp32 GEMM) -> BN -> ReLU -> 27-pt channel-
// wise sparse conv (gather) } -> sum -> BN -> +residual -> ReLU
// N=100000 voxels, C=128 channels, K=27 offsets, M=4 branches. fp32.
// ---------------------------------------------------------------------------

#define CCH   128
#define KOFF  27
#define MBR   4
#define EPSV  1e-5f
#define LDSX  132   // padded LDS row stride (floats): 132%64=4 banks/row ->
                    // 16 consecutive rows touch distinct bank pairs for b64

typedef __attribute__((ext_vector_type(2))) float v2f;
typedef __attribute__((ext_vector_type(8))) float v8f;

// ---------------------------------------------------------------------------
// Kernel 0: zero the stats scratch ((M+1) slots of [sum[128] | sumsq[128]])
// ---------------------------------------------------------------------------
__global__ void zero_stats_kernel(float* __restrict__ stats, int n) {
    for (int i = threadIdx.x; i < n; i += blockDim.x) stats[i] = 0.0f;
}

// ---------------------------------------------------------------------------
// Kernel 1: t = x @ W  (N x 128 @ 128 x 128), fp32 WMMA 16x16x4, fused
// per-column sum / sum-of-squares for training-mode BN.
//
// Block = 256 threads = 8 waves; block tile = 32 rows x 128 cols.
// The 32x128 x-tile is staged in LDS once (deduplicates A-fragment loads
// across the 8 waves); wave w owns columns [16w,16w+16) and computes TWO
// 16x16 output tiles (rows r0..r0+15 and r0+16..r0+31), reusing each B
// fragment for two WMMAs.
//
// VGPR layouts (ISA 7.12.2):
//   A (16x4 f32): both half-waves hold rows M=0..15; VGPR0=K{0,2}, VGPR1=K{1,3}.
//   B (4x16 f32): lane n in 0..15 holds col n; VGPR0=K{0,2}, VGPR1=K{1,3}.
//   C/D (16x16 f32): VGPR i -> row (8*hi + i), col = lane&15.
// ---------------------------------------------------------------------------
__global__ __launch_bounds__(256)
void gemm_bn_stats_kernel(const float* __restrict__ x,
                          const float* __restrict__ W,    // one branch: 128x128
                          float* __restrict__ t,
                          float* __restrict__ col_sum,
                          float* __restrict__ col_sumsq)
{
    __shared__ float lx[32 * LDSX];

    const int tid = threadIdx.x;
    const int r0  = blockIdx.x * 32;

    // Stage x[r0..r0+31][0..127] -> LDS (1024 float4, 4 per thread, coalesced)
    {
        const float4* __restrict__ src = (const float4*)(x + (size_t)r0 * CCH);
        for (int i = tid; i < 32 * 32; i += 256) {
            const int row = i >> 5;
            const int c4  = i & 31;
            *(float4*)(&lx[row * LDSX + c4 * 4]) = src[row * 32 + c4];
        }
    }
    __syncthreads();

    const int lane = tid & 31;
    const int wave = tid >> 5;              // 0..7 -> column tile
    const int lo   = lane & 15;
    const int hi   = lane >> 4;             // 0 or 1
    const int c0   = wave * 16;

    v8f acc0 = {};
    v8f acc1 = {};
#pragma unroll
    for (int kk = 0; kk < CCH; kk += 4) {
        const int ka = kk + hi * 2;
        // A fragments from LDS (bank-conflict-free b64 reads)
        v2f a0 = *(const v2f*)(&lx[lo * LDSX + ka]);
        v2f a1 = *(const v2f*)(&lx[(16 + lo) * LDSX + ka]);
        // B fragment: W[ka][c0+lo], W[ka+1][c0+lo] (reused by both WMMAs)
        v2f b;
        b.x = W[(size_t)ka * CCH + c0 + lo];
        b.y = W[(size_t)(ka + 1) * CCH + c0 + lo];
        acc0 = __builtin_amdgcn_wmma_f32_16x16x4_f32(
            false, a0, false, b, (short)0, acc0, false, false);
        acc1 = __builtin_amdgcn_wmma_f32_16x16x4_f32(
            false, a1, false, b, (short)0, acc1, false, false);
    }

    // Epilogue: scatter both tiles to t; fold 16 rows of this column into
    // the BN statistics, then one atomic pair per lane.
    float s = 0.0f, sq = 0.0f;
#pragma unroll
    for (int i = 0; i < 8; ++i) {
        const float v0 = acc0[i];
        const float v1 = acc1[i];
        t[(size_t)(r0 + hi * 8 + i) * CCH + c0 + lo]      = v0;
        t[(size_t)(r0 + 16 + hi * 8 + i) * CCH + c0 + lo] = v1;
        s  += v0 + v1;
        sq += v0 * v0 + v1 * v1;
    }
    atomicAdd(&col_sum[c0 + lo], s);
    atomicAdd(&col_sumsq[c0 + lo], sq);
}

// ---------------------------------------------------------------------------
// Kernel 2: in-place h = relu(bn(t))   (float4 vectorized elementwise)
// ---------------------------------------------------------------------------
__global__ __launch_bounds__(256)
void bn_relu_kernel(float* __restrict__ t,
                    const float* __restrict__ col_sum,
                    const float* __restrict__ col_sumsq,
                    const float* __restrict__ g,
                    const float* __restrict__ b,
                    float inv_n)
{
    const size_t f = (size_t)blockIdx.x * blockDim.x + threadIdx.x; // float4 idx
    const int c = ((int)(f & 31)) * 4;                              // channel base
    float4 v  = *(const float4*)(t + f * 4);
    float4 sm = *(const float4*)(col_sum + c);
    float4 sg = *(const float4*)(col_sumsq + c);
    float4 gg = *(const float4*)(g + c);
    float4 bb = *(const float4*)(b + c);

    float mu, var, rs;
    mu = sm.x * inv_n; var = sg.x * inv_n - mu * mu; rs = rsqrtf(var + EPSV);
    v.x = fmaxf(0.0f, (v.x - mu) * rs * gg.x + bb.x);
    mu = sm.y * inv_n; var = sg.y * inv_n - mu * mu; rs = rsqrtf(var + EPSV);
    v.y = fmaxf(0.0f, (v.y - mu) * rs * gg.y + bb.y);
    mu = sm.z * inv_n; var = sg.z * inv_n - mu * mu; rs = rsqrtf(var + EPSV);
    v.z = fmaxf(0.0f, (v.z - mu) * rs * gg.z + bb.z);
    mu = sm.w * inv_n; var = sg.w * inv_n - mu * mu; rs = rsqrtf(var + EPSV);
    v.w = fmaxf(0.0f, (v.w - mu) * rs * gg.w + bb.w);

    *(float4*)(t + f * 4) = v;
}

// ---------------------------------------------------------------------------
// Kernel 3: gather-accumulate   out[n,:] (+)= sum_k h[nbr[n,k],:] * wk[k,:]
// block = 256 threads = 8 waves; block handles 16 voxels (2 per wave);
// wk[m] (27x128 f32 = 13.8 KB) staged in LDS; per-lane float4 channel slice
// -> each gather is a fully coalesced 512B row read served from L2.
// ---------------------------------------------------------------------------
__global__ __launch_bounds__(256)
void gather_acc_kernel(const float* __restrict__ h,
                       const int*   __restrict__ nbr,
                       const float* __restrict__ wkm,   // one branch: 27x128
                       float* __restrict__ out,
                       int beta)                        // 0: write, 1: accumulate
{
    __shared__ float lwk[KOFF * CCH];
    for (int i = threadIdx.x; i < KOFF * CCH; i += 256) lwk[i] = wkm[i];
    __syncthreads();

    const int wave = threadIdx.x >> 5;
    const int lane = threadIdx.x & 31;
    const int c    = lane * 4;
    const int vbase = blockIdx.x * 16;

#pragma unroll
    for (int i = 0; i < 2; ++i) {
        const int v = vbase + wave * 2 + i;
        const int* __restrict__ nb = nbr + (size_t)v * KOFF;
        float4 acc = make_float4(0.f, 0.f, 0.f, 0.f);
#pragma unroll
        for (int k = 0; k < KOFF; ++k) {
            const int idx = nb[k];                       // wave-uniform load
            const float4 hv = *(const float4*)(h + (size_t)idx * CCH + c);
            const float4 w  = *(const float4*)(&lwk[k * CCH + c]);
            acc.x += hv.x * w.x; acc.y += hv.y * w.y;
            acc.z += hv.z * w.z; acc.w += hv.w * w.w;
        }
        float4* op = (float4*)(out + (size_t)v * CCH + c);
        if (beta) {
            float4 o = *op;
            o.x += acc.x; o.y += acc.y; o.z += acc.z; o.w += acc.w;
            *op = o;
        } else {
            *op = acc;
        }
    }
}

// ---------------------------------------------------------------------------
// Kernel 4: column sums / sum-of-squares of the accumulated output
// (for the final training-mode BN). 128 threads = one column each.
// ---------------------------------------------------------------------------
__global__ __launch_bounds__(128)
void col_stats_kernel(const float* __restrict__ a,
                      float* __restrict__ col_sum,
                      float* __restrict__ col_sumsq,
                      int nrows)
{
    const int c = threadIdx.x;
    float s = 0.0f, sq = 0.0f;
    for (int r = blockIdx.x; r < nrows; r += gridDim.x) {
        const float v = a[(size_t)r * CCH + c];
        s += v; sq += v * v;
    }
    atomicAdd(&col_sum[c], s);
    atomicAdd(&col_sumsq[c], sq);
}

// ---------------------------------------------------------------------------
// Kernel 5: out = relu( relu(bn(acc)) + x )   in place over d_out
// ---------------------------------------------------------------------------
__global__ __launch_bounds__(256)
void final_bn_res_relu_kernel(float* __restrict__ out,       // holds acc
                              const float* __restrict__ x,
                              const float* __restrict__ col_sum,
                              const float* __restrict__ col_sumsq,
                              const float* __restrict__ g,
                              const float* __restrict__ b,
                              float inv_n)
{
    const size_t f = (size_t)blockIdx.x * blockDim.x + threadIdx.x; // float4 idx
    const int c = ((int)(f & 31)) * 4;
    float4 v  = *(const float4*)(out + f * 4);
    float4 xr = *(const float4*)(x + f * 4);
    float4 sm = *(const float4*)(col_sum + c);
    float4 sg = *(const float4*)(col_sumsq + c);
    float4 gg = *(const float4*)(g + c);
    float4 bb = *(const float4*)(b + c);

    float mu, var, rs;
    mu = sm.x * inv_n; var = sg.x * inv_n - mu * mu; rs = rsqrtf(var + EPSV);
    v.x = fmaxf(0.0f, fmaxf(0.0f, (v.x - mu) * rs * gg.x + bb.x) + xr.x);
    mu = sm.y * inv_n; var = sg.y * inv_n - mu * mu; rs = rsqrtf(var + EPSV);
    v.y = fmaxf(0.0f, fmaxf(0.0f, (v.y - mu) * rs * gg.y + bb.y) + xr.y);
    mu = sm.z * inv_n; var = sg.z * inv_n - mu * mu; rs = rsqrtf(var + EPSV);
    v.z = fmaxf(0.0f, fmaxf(0.0f, (v.z - mu) * rs * gg.z + bb.z) + xr.z);
    mu = sm.w * inv_n; var = sg.w * inv_n - mu * mu; rs = rsqrtf(var + EPSV);
    v.w = fmaxf(0.0f, fmaxf(0.0f, (v.w - mu) * rs * gg.w + bb.w) + xr.w);

    *(float4*)(out + f * 4) = v;
}

// ---------------------------------------------------------------------------
// Host-side launch sequence (graph-capture safe: kernels only, all on stream)
// ---------------------------------------------------------------------------
extern "C" void kernel_launch(void* const* d_in, const int* in_sizes, int n_in,
                              void* d_out, int out_size, void* d_ws, size_t ws_size,
                              hipStream_t stream) {
    const float* x     = (const float*)d_in[0];   // [N,128]
    const int*   nbr   = (const int*)  d_in[1];   // [N,27]
    const float* W1    = (const float*)d_in[2];   // [4,128,128]
    const float* g1    = (const float*)d_in[3];   // [4,128]
    const float* b1    = (const float*)d_in[4];   // [4,128]
    const float* wk    = (const float*)d_in[5];   // [4,27,128]
    const float* g_out = (const float*)d_in[6];   // [128]
    const float* b_out = (const float*)d_in[7];   // [128]
    float* out = (float*)d_out;

    const int N = in_sizes[0] / CCH;              // 100000 (divisible by 32)
    const float inv_n = 1.0f / (float)N;

    // Workspace layout: h[N*128] fp32, then (M+1) stat slots of 256 floats.
    float* h     = (float*)d_ws;
    float* stats = h + (size_t)N * CCH;
    const int n_stats = (MBR + 1) * 2 * CCH;

    zero_stats_kernel<<<1, 256, 0, stream>>>(stats, n_stats);

    const int gemm_blocks   = N / 32;             // 3125
    const int gather_blocks = N / 16;             // 6250
    const int ew_blocks     = (N * CCH) / (4 * 256); // 12500 (exact)

    for (int m = 0; m < MBR; ++m) {
        float* ssum = stats + m * 2 * CCH;
        float* ssq  = ssum + CCH;
        // Conv1x1 via fp32 WMMA + fused BN statistics
        gemm_bn_stats_kernel<<<gemm_blocks, 256, 0, stream>>>(
            x, W1 + (size_t)m * CCH * CCH, h, ssum, ssq);
        // h = relu(bn(t)) in place
        bn_relu_kernel<<<ew_blocks, 256, 0, stream>>>(
            h, ssum, ssq, g1 + m * CCH, b1 + m * CCH, inv_n);
        // out (+)= sum_k h[nbr[:,k]] * wk[m,k]
        gather_acc_kernel<<<gather_blocks, 256, 0, stream>>>(
            h, nbr, wk + (size_t)m * KOFF * CCH, out, m > 0 ? 1 : 0);
    }

    // Final BN stats over accumulated out, then BN + residual + ReLU in place.
    float* osum = stats + MBR * 2 * CCH;
    float* osq  = osum + CCH;
    col_stats_kernel<<<1024, 128, 0, stream>>>(out, osum, osq, N);
    final_bn_res_relu_kernel<<<ew_blocks, 256, 0, stream>>>(
        out, x, osum, osq, g_out, b_out, inv_n);
}